// PSLFLevel_22978075034039
// MI455X (gfx1250) — compile-verified
//
#include <hip/hip_runtime.h>
#include <hip/hip_bf16.h>
#include <math.h>

typedef float v2f __attribute__((ext_vector_type(2)));
typedef float v8f __attribute__((ext_vector_type(8)));

// Problem constants
#define BB      8
#define CC      64
#define HH      256
#define WW      256
#define H2      128
#define W2      128
#define ORIENTN 4
#define CHSTR   (HH * WW)      // x channel stride (65536 floats)

// Tiling: each workgroup produces a (R2T x U2T) output tile for all 64 channels.
#define R2T   2                // output rows per WG
#define U2T   4                // output cols per WG
#define TROWS 12               // 2*R2T + 8 staged y rows (vertical halo +-2 in decimated grid)
#define TCOLS 16               // 2*U2T + 8 staged y cols (horizontal halo)
#define RBLK  (H2 / R2T)       // 64
#define CBLK  (W2 / U2T)       // 32

// Workspace layout (floats)
#define WS_Q    0              // q[64][64]
#define WS_COEF 4096           // 8 arrays x 64 per-channel cos/sin
#define WS_OR   4608           // 4x cos, 4x sin, 4x tanh(diag), 4x softplus(gain)

// Output offsets (floats): ll | detail | orientation_field
#define LL_OFF  0L
#define DET_OFF 8388608L       // 8*64*128*128
#define ORI_OFF 41943040L      // + 8*256*128*128

__global__ void pslf_setup(const float* __restrict__ pvec,
                           const float* __restrict__ pv, const float* __restrict__ uv,
                           const float* __restrict__ ph, const float* __restrict__ uh,
                           const float* __restrict__ ang, const float* __restrict__ dw,
                           const float* __restrict__ sg, float* __restrict__ ws) {
    const int c = threadIdx.x;  // 64 threads
    float s1 = 0.f, s2 = 0.f, d = 0.f;
    for (int i = 0; i < 64; ++i) {
        const float a = pvec[i], b = pvec[64 + i];
        s1 += a * a; s2 += b * b; d += a * b;
    }
    const float inv1 = rsqrtf(fmaxf(s1, 1e-6f));
    const float inv2 = rsqrtf(fmaxf(s2, 1e-6f));
    const float dot  = d * inv1 * inv2;   // u2 . u1
    const float u1c = pvec[c] * inv1;
    const float u2c = pvec[64 + c] * inv2;
    // q = (I - 2 u2 u2^T)(I - 2 u1 u1^T)
    for (int o = 0; o < 64; ++o) {
        const float u1o = pvec[o] * inv1;
        const float u2o = pvec[64 + o] * inv2;
        const float q = ((o == c) ? 1.f : 0.f)
                      - 2.f * u1o * u1c - 2.f * u2o * u2c
                      + 4.f * dot * u2o * u1c;
        ws[WS_Q + o * 64 + c] = q;
    }
    ws[WS_COEF + 0 * 64 + c] = cosf(pv[c]);
    ws[WS_COEF + 1 * 64 + c] = sinf(pv[c]);
    ws[WS_COEF + 2 * 64 + c] = cosf(uv[c]);
    ws[WS_COEF + 3 * 64 + c] = sinf(uv[c]);
    ws[WS_COEF + 4 * 64 + c] = cosf(ph[c]);
    ws[WS_COEF + 5 * 64 + c] = sinf(ph[c]);
    ws[WS_COEF + 6 * 64 + c] = cosf(uh[c]);
    ws[WS_COEF + 7 * 64 + c] = sinf(uh[c]);
    if (c < ORIENTN) {
        ws[WS_OR + c]      = cosf(ang[c]);
        ws[WS_OR + 4 + c]  = sinf(ang[c]);
        ws[WS_OR + 8 + c]  = tanhf(dw[c]);
        const float g = sg[c];  // scale_gains[0][c], row-major (3,8)
        ws[WS_OR + 12 + c] = (g > 0.f) ? (g + log1pf(expf(-g))) : log1pf(expf(g));
    }
}

__global__ __launch_bounds__(256)
void pslf_main(const float* __restrict__ x, const float* __restrict__ ws,
               float* __restrict__ out) {
    __shared__ float yl[CC * TROWS * TCOLS];       // 48 KB staged channel-mixed tile
    __shared__ float oacc[ORIENTN * R2T * U2T];    // orientation channel-mean accumulator

    const int bid = blockIdx.x;
    const int cb = bid % CBLK;
    const int rb = (bid / CBLK) % RBLK;
    const int b  = bid / (CBLK * RBLK);
    const int r0 = rb * R2T;            // output row base
    const int u0 = cb * U2T;            // output col base
    const int ybase = 2 * r0 - 4;       // first staged y row (zero-padded outside)
    const int wbase = 2 * u0 - 4;       // first staged y col

    const int tid = threadIdx.x;
    if (tid < ORIENTN * R2T * U2T) oacc[tid] = 0.0f;

    // ---------------- Stage 1: y = q @ x via V_WMMA_F32_16X16X4_F32 ----------------
    const int wid   = tid >> 5;
    const int lane  = tid & 31;
    const int hf    = lane >> 4;        // half-wave index
    const int m     = lane & 15;        // M row (and N col) within 16x16 tile
    const int grp   = wid & 3;          // output-channel group (4 groups of 16)
    const int mb    = grp << 4;
    const int rhalf = wid >> 2;         // row split across wave pairs

    // Preload A (q sub-block) per ISA 16x4 f32 A layout:
    // lanes 0-15: M=0..15, VGPR0=K0, VGPR1=K1; lanes 16-31: K2/K3.
    v2f A[16];
#pragma unroll
    for (int s = 0; s < 16; ++s) {
        const float* qrow = ws + WS_Q + (mb + m) * 64 + 4 * s + 2 * hf;
        v2f a; a.x = qrow[0]; a.y = qrow[1];
        A[s] = a;
    }

    const int  w    = wbase + m;
    const int  wc   = min(max(w, 0), WW - 1);       // clamped (loads always in-bounds)
    const bool okw  = (w >= 0) & (w < WW);
    const float* xbp = x + (long)b * CC * CHSTR;    // uniform base -> SADDR addressing

    for (int ii = 0; ii < 6; ++ii) {
        const int  i   = rhalf * 6 + ii;            // staged tile row
        const int  yr  = ybase + i;                 // global y row
        const bool ok  = okw & (yr >= 0) & (yr < HH);
        const int  yrc = min(max(yr, 0), HH - 1);   // clamped
        const int  off = yrc * WW + wc;             // 32-bit per-lane offset
        if (ii + 1 < 6 && (yr + 1) >= 0 && (yr + 1) < HH)
            __builtin_prefetch(xbp + off + WW, 0, 1);   // global_prefetch_b8

        v8f acc = {0.f, 0.f, 0.f, 0.f, 0.f, 0.f, 0.f, 0.f};
#pragma unroll
        for (int s = 0; s < 16; ++s) {
            const int c0 = 4 * s + 2 * hf;          // B layout mirrors A (K rows)
            v2f bb;                                 // unconditional, clamped loads
            bb.x = xbp[off + c0 * CHSTR];
            bb.y = xbp[off + (c0 + 1) * CHSTR];
            acc = __builtin_amdgcn_wmma_f32_16x16x4_f32(
                      false, A[s], false, bb, (short)0, acc, false, false);
        }
        // D layout: VGPR v -> M = v + 8*hf, N = m.  Zero-pad OOB pixels here
        // (store lane's n == its B lane's n, so the mask is consistent).
#pragma unroll
        for (int v = 0; v < 8; ++v)
            yl[(mb + v + 8 * hf) * (TROWS * TCOLS) + i * TCOLS + m] = ok ? acc[v] : 0.0f;
    }

    __syncthreads();

    // Orientation coefficients (uniform)
    float ock[4], osk[4], odk[4], ogk[4];
#pragma unroll
    for (int k = 0; k < 4; ++k) {
        ock[k] = ws[WS_OR + k];
        osk[k] = ws[WS_OR + 4 + k];
        odk[k] = ws[WS_OR + 8 + k];
        ogk[k] = ws[WS_OR + 12 + k];
    }

    // ---------------- Stage 2: lifting + orientation ----------------
    // 512 work items (64 ch x 2 rows x 4 cols), 2 per thread.
    for (int it = 0; it < 2; ++it) {
        const int idx = tid + 256 * it;
        const int lu = idx & 3;
        const int lr = (idx >> 2) & 1;
        const int c  = idx >> 3;
        const float cvp = ws[WS_COEF + 0 * 64 + c];
        const float svp = ws[WS_COEF + 1 * 64 + c];
        const float cvu = ws[WS_COEF + 2 * 64 + c];
        const float svu = ws[WS_COEF + 3 * 64 + c];
        const float chp = ws[WS_COEF + 4 * 64 + c];
        const float shp = ws[WS_COEF + 5 * 64 + c];
        const float chu = ws[WS_COEF + 6 * 64 + c];
        const float shu = ws[WS_COEF + 7 * 64 + c];
        const float* yc = yl + c * (TROWS * TCOLS);

        // even[r0+a] -> tile row 2a+4 ; odd[r0+a] -> tile row 2a+5
        float lowv[9], detc[9];
#pragma unroll
        for (int t = 0; t < 9; ++t) {
            const int j = 2 * lu + t;
            const float e_m2 = yc[(2 * lr + 0) * TCOLS + j];
            const float e_m1 = yc[(2 * lr + 2) * TCOLS + j];
            const float e_0  = yc[(2 * lr + 4) * TCOLS + j];
            const float e_p1 = yc[(2 * lr + 6) * TCOLS + j];
            const float e_p2 = yc[(2 * lr + 8) * TCOLS + j];
            const float o_m1 = yc[(2 * lr + 3) * TCOLS + j];
            const float o_0  = yc[(2 * lr + 5) * TCOLS + j];
            const float o_p1 = yc[(2 * lr + 7) * TCOLS + j];
            const float dm = o_m1 - cvp * e_m2 - svp * e_0;
            const float d0 = o_0  - cvp * e_m1 - svp * e_p1;
            const float dp = o_p1 - cvp * e_0  - svp * e_p2;
            detc[t] = d0;
            lowv[t] = e_0 + cvu * dm + svu * dp;
        }

        const float lh_m = lowv[3] - chp * lowv[0] - shp * lowv[4];
        const float lh_0 = lowv[5] - chp * lowv[2] - shp * lowv[6];
        const float lh_p = lowv[7] - chp * lowv[4] - shp * lowv[8];
        const float llv  = lowv[4] + chu * lh_m + shu * lh_p;
        const float hh_m = detc[3] - chp * detc[0] - shp * detc[4];
        const float hh_0 = detc[5] - chp * detc[2] - shp * detc[6];
        const float hh_p = detc[7] - chp * detc[4] - shp * detc[8];
        const float hlv  = detc[4] + chu * hh_m + shu * hh_p;

        const int r = r0 + lr, u = u0 + lu;
        out[LL_OFF + (((long)b * CC + c) * H2 + r) * W2 + u] = llv;
#pragma unroll
        for (int k = 0; k < 4; ++k) {
            const float st = (ock[k] * hlv + osk[k] * lh_0 + odk[k] * hh_0) * ogk[k];
            out[DET_OFF + (((long)b * (4 * CC) + k * CC + c) * H2 + r) * W2 + u] = st;
            atomicAdd(&oacc[(k * R2T + lr) * U2T + lu], st * (1.0f / 64.0f));
        }
    }

    __syncthreads();

    if (tid < ORIENTN * R2T * U2T) {
        const int lu = tid & 3;
        const int lr = (tid >> 2) & 1;
        const int k  = tid >> 3;
        out[ORI_OFF + (((long)b * ORIENTN + k) * H2 + (r0 + lr)) * W2 + (u0 + lu)] = oacc[tid];
    }
}

extern "C" void kernel_launch(void* const* d_in, const int* in_sizes, int n_in,
                              void* d_out, int out_size, void* d_ws, size_t ws_size,
                              hipStream_t stream) {
    (void)in_sizes; (void)n_in; (void)out_size; (void)ws_size;
    const float* x    = (const float*)d_in[0];
    const float* pvec = (const float*)d_in[1];
    const float* pv   = (const float*)d_in[2];
    const float* uv   = (const float*)d_in[3];
    const float* ph   = (const float*)d_in[4];
    const float* uh   = (const float*)d_in[5];
    const float* ang  = (const float*)d_in[6];
    const float* dw   = (const float*)d_in[7];
    const float* sg   = (const float*)d_in[8];
    float* ws  = (float*)d_ws;
    float* out = (float*)d_out;

    pslf_setup<<<1, 64, 0, stream>>>(pvec, pv, uv, ph, uh, ang, dw, sg, ws);
    pslf_main<<<dim3(BB * RBLK * CBLK), dim3(256), 0, stream>>>(x, ws, out);
}